// long_range_module_sph_78185584657109
// MI455X (gfx1250) — compile-verified
//
#include <hip/hip_runtime.h>
#include <hip/hip_bf16.h>
#include <math.h>

#define NN   50000
#define GG   1000
#define EE   200000
#define CC   128
#define NRBF 32
#define FCOLS 640   // 5*C

typedef float v2f __attribute__((ext_vector_type(2)));
typedef float v8f __attribute__((ext_vector_type(8)));

// ---------------------------------------------------------------- utilities
__global__ void zero_kernel(float* __restrict__ p, int n) {
  int i = blockIdx.x * blockDim.x + threadIdx.x;
  if (i < n) p[i] = 0.0f;
}

// ------------------------------------------------- group statistics (stage 1)
__global__ void group_stats_kernel(const float* __restrict__ pos,
                                   const float* __restrict__ az,
                                   const int*   __restrict__ labels,
                                   float* __restrict__ zsum,
                                   float* __restrict__ cnt,
                                   float* __restrict__ wpos) {
  int n = blockIdx.x * blockDim.x + threadIdx.x;
  if (n >= NN) return;
  int g = labels[n];
  float a = az[n];
  atomicAdd(&zsum[g], a);
  atomicAdd(&cnt[g], 1.0f);
  atomicAdd(&wpos[g * 3 + 0], pos[n * 3 + 0] * a);
  atomicAdd(&wpos[g * 3 + 1], pos[n * 3 + 1] * a);
  atomicAdd(&wpos[g * 3 + 2], pos[n * 3 + 2] * a);
}

__global__ void group_pos_finalize(float* __restrict__ wpos,
                                   const float* __restrict__ zsum) {
  int i = blockIdx.x * blockDim.x + threadIdx.x;
  if (i >= GG * 3) return;
  wpos[i] = wpos[i] / zsum[i / 3];
}

// ------------------------------------------- per-edge geometry + RBF (stage 2)
__global__ void edge_geom_kernel(const float* __restrict__ pos,
                                 const float* __restrict__ gpos,
                                 const int*   __restrict__ node_id,
                                 const int*   __restrict__ group_id,
                                 float* __restrict__ sh,
                                 float* __restrict__ rbf) {
  int e = blockIdx.x * blockDim.x + threadIdx.x;
  if (e >= EE) return;
  int n = node_id[e], g = group_id[e];
  float vx = pos[n * 3 + 0] - gpos[g * 3 + 0];
  float vy = pos[n * 3 + 1] - gpos[g * 3 + 1];
  float vz = pos[n * 3 + 2] - gpos[g * 3 + 2];
  float d = sqrtf(vx * vx + vy * vy + vz * vz);
  float inv = 1.0f / d;
  float hx = vx * inv, hy = vy * inv, hz = vz * inv;
  const float SQ3 = 1.7320508075688772f;
  // p = vhat[[1,2,0]]; sh1 = sqrt(3)*p
  sh[e * 3 + 0] = SQ3 * hy;
  sh[e * 3 + 1] = SQ3 * hz;
  sh[e * 3 + 2] = SQ3 * hx;
  const float step = 12.0f / 31.0f;                 // CUTOFF/(NUM_RBF-1)
  const float gamma = (31.0f / 12.0f) * (31.0f / 12.0f);
  float* rr = rbf + (size_t)e * NRBF;
#pragma unroll
  for (int i = 0; i < NRBF; ++i) {
    float t = d - (float)i * step;
    rr[i] = expf(-gamma * t * t);
  }
}

// ------------------------------------------------- segment mean of node_attr
__global__ void ge_accum_kernel(const float* __restrict__ na,
                                const int*   __restrict__ labels,
                                float* __restrict__ ge) {
  int i = blockIdx.x * blockDim.x + threadIdx.x;   // N*512 threads
  if (i >= NN * 512) return;
  int n = i >> 9;
  int c = i & 511;
  atomicAdd(&ge[labels[n] * 512 + c], na[i]);
}

__global__ void ge_div_kernel(float* __restrict__ ge,
                              const float* __restrict__ cnt) {
  int i = blockIdx.x * blockDim.x + threadIdx.x;   // G*512 threads
  if (i >= GG * 512) return;
  ge[i] = ge[i] / cnt[i >> 9];
}

// ---------------------------------------------------------------------------
// Fused edge kernel: WMMA computes w0..w4 = (rbf_tile @ fc_w)/sqrt(32) for a
// 16-edge x 16-channel tile, then the spherical tensor-product epilogue and
// scatter-add into node_attr.
// One wave per tile; grid is exact so EXEC==all-ones for every WMMA.
// ---------------------------------------------------------------------------
__launch_bounds__(256)
__global__ void edge_wmma_kernel(const float* __restrict__ rbf,
                                 const float* __restrict__ fcw,
                                 const float* __restrict__ ge,
                                 const float* __restrict__ sh,
                                 const int*   __restrict__ node_id,
                                 const int*   __restrict__ group_id,
                                 float* __restrict__ na) {
  const int lane  = threadIdx.x & 31;
  const int wave  = (blockIdx.x * blockDim.x + threadIdx.x) >> 5;
  const int etile = wave >> 3;          // E/16 = 12500 edge tiles
  const int ct    = wave & 7;           // C/16 = 8 channel tiles
  const int c0    = ct << 4;
  const int row   = lane & 15;
  const int hi    = lane >> 4;
  const int ebase = etile << 4;

  // A matrix (16x32 f32) in 16x16x4-step layout:
  // lanes 0-15 hold K = 4s+0,4s+1 ; lanes 16-31 hold K = 4s+2,4s+3
  v2f a[8];
  const float* ar = rbf + (size_t)(ebase + row) * NRBF + 2 * hi;
#pragma unroll
  for (int s = 0; s < 8; ++s) {
    a[s].x = ar[4 * s + 0];
    a[s].y = ar[4 * s + 1];
  }

  // Five 16x16 accumulators: w_j = A @ fc_w[:, j*128 + c0 : +16]
  v8f acc[5];
#pragma unroll
  for (int j = 0; j < 5; ++j) {
    v8f c = {0.f, 0.f, 0.f, 0.f, 0.f, 0.f, 0.f, 0.f};
    const float* bcol = fcw + j * CC + c0 + row + 2 * hi * FCOLS;
#pragma unroll
    for (int s = 0; s < 8; ++s) {
      v2f b;
      b.x = bcol[(4 * s + 0) * FCOLS];
      b.y = bcol[(4 * s + 1) * FCOLS];
      c = __builtin_amdgcn_wmma_f32_16x16x4_f32(false, a[s], false, b,
                                                (short)0, c, false, false);
    }
    acc[j] = c;
  }

  // coefficients with 1/sqrt(NUM_RBF) folded in
  constexpr float INVS = 0.17677669529663687f;      // 1/sqrt(32)
  constexpr float SQ3  = 1.7320508075688772f;
  constexpr float SQ6  = 2.449489742783178f;
  constexpr float S15  = 0.4472135954999579f;       // sqrt(1/5)
  constexpr float S35  = 0.7745966692414834f;       // sqrt(3/5)
  const float k0c = S15 * INVS;
  const float k3c = (S15 / SQ3) * INVS;
  const float k1c = (S35 / SQ3) * INVS;
  const float k2c = (S35 / SQ3) * INVS;
  const float k4c = (S35 / SQ6) * INVS;

  const int ch = c0 + row;                          // channel handled by lane
#pragma unroll
  for (int r = 0; r < 8; ++r) {
    const int e  = ebase + r + 8 * hi;              // C-layout: M = r + 8*hi
    const int g  = group_id[e];
    const int nd = node_id[e];
    const float* xg = ge + (size_t)g * 512;
    float x10 = xg[ch];
    float xa  = xg[128 + 3 * ch + 0];               // x1_1[e,c,i] = x1[C+3c+i]
    float xb  = xg[128 + 3 * ch + 1];
    float xc  = xg[128 + 3 * ch + 2];
    float s0 = sh[e * 3 + 0], s1 = sh[e * 3 + 1], s2 = sh[e * 3 + 2];
    float w0 = acc[0][r], w1 = acc[1][r], w2 = acc[2][r];
    float w3 = acc[3][r], w4 = acc[4][r];

    float dot = xa * s0 + xb * s1 + xc * s2;
    float o0  = k0c * w0 * x10 + k3c * w3 * dot;
    float cr0 = xb * s2 - xc * s1;                  // cross(x1_1, sh1)
    float cr1 = xc * s0 - xa * s2;
    float cr2 = xa * s1 - xb * s0;
    float t1  = k1c * w1 * x10;
    float oa  = t1 * s0 + k2c * w2 * xa + k4c * w4 * cr0;
    float ob  = t1 * s1 + k2c * w2 * xb + k4c * w4 * cr1;
    float oc  = t1 * s2 + k2c * w2 * xc + k4c * w4 * cr2;

    float* dst = na + (size_t)nd * 512;
    atomicAdd(dst + ch, o0);
    atomicAdd(dst + 128 + 3 * ch + 0, oa);
    atomicAdd(dst + 128 + 3 * ch + 1, ob);
    atomicAdd(dst + 128 + 3 * ch + 2, oc);
  }
}

// --------------------------------------------- per-node normalization (wave32)
__global__ void norm_kernel(float* __restrict__ na,
                            const float* __restrict__ w0,
                            const float* __restrict__ b0,
                            const float* __restrict__ w1) {
  const int lane = threadIdx.x & 31;
  const int node = (blockIdx.x * blockDim.x + threadIdx.x) >> 5;
  if (node >= NN) return;
  float* f = na + (size_t)node * 512;

  // f0 layernorm over 128 channels (4 per lane)
  float4 v = ((float4*)f)[lane];
  float s = v.x + v.y + v.z + v.w;
#pragma unroll
  for (int m = 16; m >= 1; m >>= 1) s += __shfl_xor(s, m, 32);
  float mean = s * (1.0f / 128.0f);
  float dx = v.x - mean, dy = v.y - mean, dz = v.z - mean, dw = v.w - mean;
  float ss = dx * dx + dy * dy + dz * dz + dw * dw;
#pragma unroll
  for (int m = 16; m >= 1; m >>= 1) ss += __shfl_xor(ss, m, 32);
  float inv = rsqrtf(ss * (1.0f / 128.0f) + 1e-5f);
  float4 wv = ((const float4*)w0)[lane];
  float4 bv = ((const float4*)b0)[lane];
  float4 o;
  o.x = dx * inv * wv.x + bv.x;
  o.y = dy * inv * wv.y + bv.y;
  o.z = dz * inv * wv.z + bv.z;
  o.w = dw * inv * wv.w + bv.w;
  ((float4*)f)[lane] = o;

  // f1 RMS-norm over (3,128) with per-channel weight  (f1[n,j,c] = flat[(1+j)*128+c])
  float4 r0 = ((float4*)f)[32 + lane];
  float4 r1 = ((float4*)f)[64 + lane];
  float4 r2 = ((float4*)f)[96 + lane];
  float q = r0.x * r0.x + r0.y * r0.y + r0.z * r0.z + r0.w * r0.w +
            r1.x * r1.x + r1.y * r1.y + r1.z * r1.z + r1.w * r1.w +
            r2.x * r2.x + r2.y * r2.y + r2.z * r2.z + r2.w * r2.w;
#pragma unroll
  for (int m = 16; m >= 1; m >>= 1) q += __shfl_xor(q, m, 32);
  float inv1 = rsqrtf(q * (1.0f / 384.0f) + 1e-5f);
  float4 w1v = ((const float4*)w1)[lane];
  float4 sc;
  sc.x = inv1 * w1v.x; sc.y = inv1 * w1v.y; sc.z = inv1 * w1v.z; sc.w = inv1 * w1v.w;
  r0.x *= sc.x; r0.y *= sc.y; r0.z *= sc.z; r0.w *= sc.w;
  r1.x *= sc.x; r1.y *= sc.y; r1.z *= sc.z; r1.w *= sc.w;
  r2.x *= sc.x; r2.y *= sc.y; r2.z *= sc.z; r2.w *= sc.w;
  ((float4*)f)[32 + lane] = r0;
  ((float4*)f)[64 + lane] = r1;
  ((float4*)f)[96 + lane] = r2;
}

// ---------------------------------------------------------------------------
extern "C" void kernel_launch(void* const* d_in, const int* in_sizes, int n_in,
                              void* d_out, int out_size, void* d_ws, size_t ws_size,
                              hipStream_t stream) {
  const float* pos   = (const float*)d_in[0];
  const float* az    = (const float*)d_in[1];
  const float* na_in = (const float*)d_in[2];
  const float* fc_w  = (const float*)d_in[3];
  const float* nl0w  = (const float*)d_in[4];
  const float* nl0b  = (const float*)d_in[5];
  const float* nl1w  = (const float*)d_in[6];
  const int*   labels = (const int*)d_in[7];
  const int*   ig     = (const int*)d_in[8];
  const int*   node_id  = ig;
  const int*   group_id = ig + EE;

  // workspace layout (floats): zsum[G] cnt[G] gpos[3G] ge[G*512] sh[3E] rbf[32E]
  float* ws   = (float*)d_ws;
  float* zsum = ws;
  float* cnt  = ws + GG;
  float* gpos = ws + 2 * GG;
  float* ge   = ws + 5 * GG;
  float* sh   = ge + GG * 512;
  float* rbf  = sh + 3 * EE;

  float* out = (float*)d_out;

  // node_attr evolves in place in d_out (inputs stay untouched)
  hipMemcpyAsync(out, na_in, sizeof(float) * (size_t)NN * 512,
                 hipMemcpyDeviceToDevice, stream);

  zero_kernel<<<(5 * GG + 255) / 256, 256, 0, stream>>>(zsum, 5 * GG);
  group_stats_kernel<<<(NN + 255) / 256, 256, 0, stream>>>(pos, az, labels,
                                                           zsum, cnt, gpos);
  group_pos_finalize<<<(3 * GG + 255) / 256, 256, 0, stream>>>(gpos, zsum);
  edge_geom_kernel<<<(EE + 255) / 256, 256, 0, stream>>>(pos, gpos, node_id,
                                                         group_id, sh, rbf);

  for (int layer = 0; layer < 2; ++layer) {
    zero_kernel<<<(GG * 512 + 255) / 256, 256, 0, stream>>>(ge, GG * 512);
    ge_accum_kernel<<<(NN * 512 + 255) / 256, 256, 0, stream>>>(out, labels, ge);
    ge_div_kernel<<<(GG * 512 + 255) / 256, 256, 0, stream>>>(ge, cnt);
    // exact grid: (E/16 tiles)*(8 channel tiles) waves, 8 waves/block
    edge_wmma_kernel<<<(EE / 16) * 8 / 8, 256, 0, stream>>>(rbf, fc_w, ge, sh,
                                                            node_id, group_id, out);
    norm_kernel<<<NN / 8, 256, 0, stream>>>(out, nl0w, nl0b, nl1w);
  }
}